// DepthWiseConv1d_23364622090654
// MI455X (gfx1250) — compile-verified
//
#include <hip/hip_runtime.h>
#include <cstdint>

// Depthwise Conv1d, k=3, stride=1, pad=1, fp32.
// [B=32, C=128, L=8192] -> memory-bound: ~268 MB traffic, ~11.5us floor @23.3TB/s.
// Strategy: one workgroup per (b,c) row; async-copy the whole 32KB row into LDS
// via CDNA5 GLOBAL_LOAD_ASYNC_TO_LDS_B128, stencil from LDS, coalesced B128 stores.

#define ROW_LEN   8192
#define NUM_CH    128
#define THREADS   256
#define VEC       4
#define SEGS      (ROW_LEN / (THREADS * VEC))   // 8 float4 segments per thread

// Builtin signature (from hipcc diagnostic): params are v4i pointers.
typedef int v4i __attribute__((ext_vector_type(4)));
typedef v4i __attribute__((address_space(1)))* gv4p_t;   // global src
typedef v4i __attribute__((address_space(3)))* lv4p_t;   // LDS dst

__global__ __launch_bounds__(THREADS) void DepthWiseConv1d_23364622090654_kernel(
    const float* __restrict__ in, const float* __restrict__ weight,
    const float* __restrict__ bias, float* __restrict__ out)
{
    extern __shared__ float tile[];             // ROW_LEN floats = 32 KB

    const int row = blockIdx.x;                 // row index = b*C + c
    const int ch  = row & (NUM_CH - 1);
    const int tid = threadIdx.x;
    const float* __restrict__ src = in  + (size_t)row * ROW_LEN;
    float*       __restrict__ dst = out + (size_t)row * ROW_LEN;

    // ---- Async global -> LDS copy of the whole row (CDNA5 path, ASYNCcnt) ----
#pragma unroll
    for (int i = 0; i < SEGS; ++i) {
        const int seg = i * THREADS + tid;      // float4 index within the row
#if __has_builtin(__builtin_amdgcn_global_load_async_to_lds_b128)
        __builtin_amdgcn_global_load_async_to_lds_b128(
            (gv4p_t)(src + seg * VEC), (lv4p_t)(tile + seg * VEC),
            /*offset=*/0, /*cpol=*/0);
#else
        // VDST = per-lane LDS byte address, VADDR = 64-bit global address.
        uint32_t lds_b = (uint32_t)(seg * 16);  // dynamic LDS begins at offset 0
        uint64_t g     = (uint64_t)(uintptr_t)(src + seg * VEC);
        asm volatile("global_load_async_to_lds_b128 %0, %1, off"
                     :: "v"(lds_b), "v"(g) : "memory");
#endif
    }
    asm volatile("s_wait_asynccnt 0" ::: "memory");  // my wave's copies landed in LDS
    __syncthreads();                                 // everyone's copies visible

    const float w0 = weight[ch * 3 + 0];
    const float w1 = weight[ch * 3 + 1];
    const float w2 = weight[ch * 3 + 2];
    const float bs = bias[ch];

    // ---- Stencil from LDS, coalesced float4 stores ----
#pragma unroll
    for (int i = 0; i < SEGS; ++i) {
        const int seg = i * THREADS + tid;
        const int l   = seg * VEC;
        const float4 m = reinterpret_cast<const float4*>(tile)[seg];
        const float left  = (l == 0)              ? 0.0f : tile[l - 1];
        const float right = (l + VEC == ROW_LEN)  ? 0.0f : tile[l + VEC];
        float4 o;
        // out[l] = w0*x[l-1] + w1*x[l] + w2*x[l+1] + bias
        o.x = fmaf(w0, left, fmaf(w1, m.x, fmaf(w2, m.y,   bs)));
        o.y = fmaf(w0, m.x,  fmaf(w1, m.y, fmaf(w2, m.z,   bs)));
        o.z = fmaf(w0, m.y,  fmaf(w1, m.z, fmaf(w2, m.w,   bs)));
        o.w = fmaf(w0, m.z,  fmaf(w1, m.w, fmaf(w2, right, bs)));
        reinterpret_cast<float4*>(dst)[seg] = o;
    }
}

extern "C" void kernel_launch(void* const* d_in, const int* in_sizes, int n_in,
                              void* d_out, int out_size, void* d_ws, size_t ws_size,
                              hipStream_t stream) {
    const float* in     = (const float*)d_in[0];   // [B, C, L] fp32
    const float* weight = (const float*)d_in[1];   // [C, 3]    fp32
    const float* bias   = (const float*)d_in[2];   // [C]       fp32
    float* out          = (float*)d_out;           // [B, C, L] fp32

    const int rows = in_sizes[0] / ROW_LEN;        // B*C = 4096
    const size_t lds_bytes = (size_t)ROW_LEN * sizeof(float);  // 32 KB

    DepthWiseConv1d_23364622090654_kernel<<<rows, THREADS, lds_bytes, stream>>>(
        in, weight, bias, out);
}